// GradientCASSI_41077067219546
// MI455X (gfx1250) — compile-verified
//
#include <hip/hip_runtime.h>
#include <hip/hip_bf16.h>

#define MD 512
#define LBANDS 31
#define CC (MD + LBANDS - 1)   // 542
#define RSTRIDE 544            // padded r row stride (2 zero cols)

typedef float v2f __attribute__((ext_vector_type(2)));
typedef float v8f __attribute__((ext_vector_type(8)));

// ---- order-preserving float<->uint encoding for atomic max ----
__device__ __forceinline__ unsigned encF(float f) {
    unsigned u = __float_as_uint(f);
    return (u & 0x80000000u) ? ~u : (u | 0x80000000u);
}
__device__ __forceinline__ float decF(unsigned e) {
    unsigned u = (e & 0x80000000u) ? (e & 0x7FFFFFFFu) : ~e;
    return __uint_as_float(u);
}
__device__ __forceinline__ unsigned umaxu(unsigned a, unsigned b) { return a > b ? a : b; }

// tridiagonal conv matrix T[k][l], k in [0,32), l in [0,31)
__device__ __forceinline__ float tval(int k, int l) {
    if (k >= LBANDS || l >= LBANDS) return 0.f;
    if (k == l) return 0.5f;
    if (k == l + 1 || k + 1 == l) return 0.25f;
    return 0.f;
}

// ---------------- K0: init scalar max slots ----------------
__global__ void k_init(unsigned* gmax) {
    gmax[0] = 0u;   // enc(max yn)
    gmax[1] = 0u;   // enc(max X2)
}

// ---------------- K1: bilinear 2x down + conv-L + *H -> Xm ----------------
__global__ void __launch_bounds__(256) k_xm(const float* __restrict__ X,
                                            const float* __restrict__ H,
                                            float* __restrict__ Xm) {
    int gid = blockIdx.x * 256 + threadIdx.x;   // 512*512 pixels
    int m = gid >> 9, n = gid & 511;

    const float bw[4] = {0.125f, 0.375f, 0.375f, 0.125f};
    float wr[4], wc[4];
    int rr[4], cx[4];
    float sr = 0.f, sc = 0.f;
#pragma unroll
    for (int k = 0; k < 4; ++k) {
        int r = 2 * m - 1 + k;
        int c = 2 * n - 1 + k;
        bool vr = (r >= 0) && (r < 1024);
        bool vc = (c >= 0) && (c < 1024);
        rr[k] = vr ? r : 0; wr[k] = vr ? bw[k] : 0.f; sr += wr[k];
        cx[k] = vc ? c : 0; wc[k] = vc ? bw[k] : 0.f; sc += wc[k];
    }
    float inv = 1.f / (sr * sc);   // per-dimension renorm (separable)

    float acc[LBANDS];
#pragma unroll
    for (int i = 0; i < LBANDS; ++i) acc[i] = 0.f;

#pragma unroll
    for (int a = 0; a < 4; ++a) {
#pragma unroll
        for (int b = 0; b < 4; ++b) {
            float w = wr[a] * wc[b] * inv;
            if (w != 0.f) {
                const float* p = X + ((size_t)rr[a] * 1024 + cx[b]) * LBANDS;
#pragma unroll
                for (int i = 0; i < LBANDS; ++i) acc[i] = fmaf(w, p[i], acc[i]);
            }
        }
    }

    size_t o = (size_t)gid * LBANDS;
#pragma unroll
    for (int i = 0; i < LBANDS; ++i) {
        float lf = (i > 0) ? acc[i - 1] : 0.f;
        float rt = (i < LBANDS - 1) ? acc[i + 1] : 0.f;
        float xc = 0.25f * lf + 0.5f * acc[i] + 0.25f * rt;
        Xm[o + i] = H[o + i] * xc;
    }
}

// ---------------- K2: yn[m,c] = sum_i Xm[m,c-i,i]; global max ----------------
__global__ void __launch_bounds__(256) k_yn(const float* __restrict__ Xm,
                                            float* __restrict__ yn,
                                            unsigned* __restrict__ gmax) {
    __shared__ unsigned smax[256];
    int gid = blockIdx.x * 256 + threadIdx.x;   // 512*542 = 277504 exact
    int m = gid / CC, c = gid - m * CC;
    float s = 0.f;
    int ilo = (c - (MD - 1) > 0) ? c - (MD - 1) : 0;
    int ihi = (c < LBANDS - 1) ? c : LBANDS - 1;
    const float* row = Xm + (size_t)m * MD * LBANDS;
    for (int i = ilo; i <= ihi; ++i)
        s += row[(size_t)(c - i) * LBANDS + i];
    yn[gid] = s;

    smax[threadIdx.x] = encF(s);
    __syncthreads();
    for (int st = 128; st > 0; st >>= 1) {
        if (threadIdx.x < st)
            smax[threadIdx.x] = umaxu(smax[threadIdx.x], smax[threadIdx.x + st]);
        __syncthreads();
    }
    if (threadIdx.x == 0) atomicMax(&gmax[0], smax[0]);
}

// ---------------- K3: r = yn/max - y, padded row stride 544 ----------------
__global__ void __launch_bounds__(256) k_res(const float* __restrict__ yn,
                                             const float* __restrict__ y,
                                             const unsigned* __restrict__ gmax,
                                             float* __restrict__ rbuf) {
    int gid = blockIdx.x * 256 + threadIdx.x;   // 512*544 = 278528 exact
    int m = gid / RSTRIDE, c = gid - m * RSTRIDE;
    float mx = decF(gmax[0]);
    float v = 0.f;
    if (c < CC) v = yn[m * CC + c] / mx - y[m * CC + c];
    rbuf[gid] = v;
}

// ---------------- K4: async-LDS stage + WMMA fp32 GEMM conv  X2 = H * (A x T) ----------------
// One wave32 per (m, 16-wide n tile): D(16x16) += A(16x4) x B(4x16), 8 K-steps, 2 N-tiles.
// The 48-float sliding window of r is staged into LDS via CDNA5 async global->LDS DMA.
__global__ void __launch_bounds__(128) k_x2(const float* __restrict__ rbuf,
                                            const float* __restrict__ H,
                                            float* __restrict__ out,
                                            unsigned* __restrict__ gmax) {
    __shared__ __align__(16) float stage[4][64];   // per-wave private window
    __shared__ unsigned smax[128];
    int lane = threadIdx.x & 31;
    int wave = threadIdx.x >> 5;
    int tile = blockIdx.x * 4 + wave;   // 512 m * 32 ntiles = 16384 tiles, 4096 blocks
    int m = tile >> 5;
    int n0 = (tile & 31) << 4;
    int hi = lane >> 4;                 // half-wave id (K split per ISA layout)
    int lm = lane & 15;                 // A-row / B-col within tile

    // ---- async stage: 64 floats per wave (48 used), GVS mode, b64 per lane ----
    unsigned voff   = (unsigned)(((size_t)m * RSTRIDE + n0 + 2 * lane) * sizeof(float));
    unsigned ldsoff = (unsigned)(uintptr_t)(&stage[wave][2 * lane]);
    asm volatile("global_load_async_to_lds_b64 %0, %1, %2"
                 :
                 : "v"(ldsoff), "v"(voff), "s"(rbuf)
                 : "memory");
    asm volatile("s_wait_asynccnt 0x0" ::: "memory");

    const float* win = stage[wave];

    unsigned em = 0u;
#pragma unroll
    for (int t = 0; t < 2; ++t) {
        v8f acc = {};
#pragma unroll
        for (int c0 = 0; c0 < 32; c0 += 4) {
            int ka = c0 + 2 * hi;
            v2f a, b;
            a.x = win[lm + ka];         // A[M=lm, K=ka]
            a.y = win[lm + ka + 1];     // A[M=lm, K=ka+1]
            int l = t * 16 + lm;        // B column (band index)
            b.x = tval(ka, l);          // B[K=ka, N=lm]
            b.y = tval(ka + 1, l);
            acc = __builtin_amdgcn_wmma_f32_16x16x4_f32(
                /*neg_a=*/false, a, /*neg_b=*/false, b,
                /*c_mod=*/(short)0, acc, /*reuse_a=*/false, /*reuse_b=*/false);
        }
        int l = t * 16 + lm;
        if (l < LBANDS) {
#pragma unroll
            for (int v = 0; v < 8; ++v) {
                int nn = n0 + v + 8 * hi;           // D row M = v + 8*hi
                size_t o = ((size_t)m * MD + nn) * LBANDS + l;
                float val = acc[v] * H[o];
                out[o] = val;
                em = umaxu(em, encF(val));
            }
        }
    }

    smax[threadIdx.x] = em;
    __syncthreads();
    for (int st = 64; st > 0; st >>= 1) {
        if (threadIdx.x < st)
            smax[threadIdx.x] = umaxu(smax[threadIdx.x], smax[threadIdx.x + st]);
        __syncthreads();
    }
    if (threadIdx.x == 0) atomicMax(&gmax[1], smax[0]);
}

// ---------------- K5: out /= max(X2) ----------------
__global__ void __launch_bounds__(256) k_scale(float* __restrict__ out,
                                               const unsigned* __restrict__ gmax) {
    int gid = blockIdx.x * 256 + threadIdx.x;   // 8126464 = 31744*256 exact
    float mx = decF(gmax[1]);
    out[gid] = out[gid] / mx;
}

extern "C" void kernel_launch(void* const* d_in, const int* in_sizes, int n_in,
                              void* d_out, int out_size, void* d_ws, size_t ws_size,
                              hipStream_t stream) {
    (void)in_sizes; (void)n_in; (void)out_size; (void)ws_size;
    const float* X = (const float*)d_in[0];   // (1,1024,1024,31)
    const float* y = (const float*)d_in[1];   // (1,512,542,1)
    const float* H = (const float*)d_in[2];   // (1,512,512,31,1)
    float* out = (float*)d_out;               // (1,512,512,31)

    // workspace layout (floats)
    float* Xm   = (float*)d_ws;                       // 512*512*31
    float* yn   = Xm + (size_t)MD * MD * LBANDS;      // 512*542
    float* rbuf = yn + (size_t)MD * CC;               // 512*544 (zero-padded)
    unsigned* gmax = (unsigned*)(rbuf + (size_t)MD * RSTRIDE); // 2 slots

    k_init<<<1, 1, 0, stream>>>(gmax);
    k_xm<<<(MD * MD) / 256, 256, 0, stream>>>(X, H, Xm);
    k_yn<<<(MD * CC) / 256, 256, 0, stream>>>(Xm, yn, gmax);
    k_res<<<(MD * RSTRIDE) / 256, 256, 0, stream>>>(yn, y, gmax, rbuf);
    k_x2<<<(MD * (MD / 16)) / 4, 128, 0, stream>>>(rbuf, H, out, gmax);
    k_scale<<<(MD * MD * LBANDS) / 256, 256, 0, stream>>>(out, gmax);
}